// LearnableKalman_20650202759779
// MI455X (gfx1250) — compile-verified
//
#include <hip/hip_runtime.h>
#include <math.h>

// CDNA5 (gfx1250) Kalman filter.  DIM=256, T=1024.
// All dense 256^3 matrix products via V_WMMA_F32_16X16X4_F32.
// Each wave computes a 32x32 macro-tile (4 WMMA tiles) for 2x fragment reuse.

#define N 256
#define T_STEPS 1024
#define EPSF 1e-8f

typedef float v2f __attribute__((ext_vector_type(2)));
typedef float v8f __attribute__((ext_vector_type(8)));

__device__ __forceinline__ v8f wmma4(v2f a, v2f b, v8f c) {
  return __builtin_amdgcn_wmma_f32_16x16x4_f32(
      /*neg_a=*/false, a, /*neg_b=*/false, b,
      /*c_mod=*/(short)0, c, /*reuse_a=*/false, /*reuse_b=*/false);
}

// ---------------------------------------------------------------------------
// 256x256x256 fp32 GEMM.  D = op(A)*op(B) [+Cin] [+diag];  optional D2 copy.
// Grid: 8 blocks x 256 threads = 64 waves; wave w owns 32x32 tile
// (rows (w>>3)*32.., cols (w&7)*32..).  K-loop: 64 steps of K=4.
// ---------------------------------------------------------------------------
template <int TA, int TB, bool HASC, bool HASDIAG, bool HASD2>
__global__ __launch_bounds__(256) void gemm256(
    const float* __restrict__ A, const float* __restrict__ B,
    const float* __restrict__ Cin, const float* __restrict__ diag,
    float* __restrict__ D, float* __restrict__ D2)
{
  const int lane  = threadIdx.x & 31;
  const int wave  = threadIdx.x >> 5;
  const int w     = blockIdx.x * 8 + wave;  // 0..63
  const int sm    = (w >> 3) * 32;
  const int sn    = (w & 7) * 32;
  const int l15   = lane & 15;
  const int lhi   = lane >> 4;              // 0 or 1
  const int khalf = lhi * 2;
  const int m0 = sm + l15, m1 = m0 + 16;    // A fragment rows
  const int n0 = sn + l15, n1 = n0 + 16;    // B fragment cols

  v8f acc00 = {}, acc01 = {}, acc10 = {}, acc11 = {};

#pragma unroll 4
  for (int kb = 0; kb < N; kb += 4) {
    const int k = kb + khalf;               // even -> 8B-aligned v2f loads OK
    v2f a0, a1, b0, b1;
    if constexpr (!TA) {                    // A(m,k) = A[m*N+k] : K-contiguous
      a0 = *(const v2f*)(A + (size_t)m0 * N + k);
      a1 = *(const v2f*)(A + (size_t)m1 * N + k);
    } else {                                // A(m,k) = A[k*N+m]
      a0.x = A[(size_t)k * N + m0];  a0.y = A[(size_t)(k + 1) * N + m0];
      a1.x = A[(size_t)k * N + m1];  a1.y = A[(size_t)(k + 1) * N + m1];
    }
    if constexpr (TB) {                     // B(k,n) = B[n*N+k] : K-contiguous
      b0 = *(const v2f*)(B + (size_t)n0 * N + k);
      b1 = *(const v2f*)(B + (size_t)n1 * N + k);
    } else {                                // B(k,n) = B[k*N+n]
      b0.x = B[(size_t)k * N + n0];  b0.y = B[(size_t)(k + 1) * N + n0];
      b1.x = B[(size_t)k * N + n1];  b1.y = B[(size_t)(k + 1) * N + n1];
    }
    acc00 = wmma4(a0, b0, acc00);
    acc01 = wmma4(a0, b1, acc01);
    acc10 = wmma4(a1, b0, acc10);
    acc11 = wmma4(a1, b1, acc11);
  }

  // Epilogue: straight-line per template config.
  const int tn0 = sn + l15, tn1 = sn + 16 + l15;
#pragma unroll
  for (int v = 0; v < 8; ++v) {
    const int r0 = sm + v + 8 * lhi;        // rows of acc0x
    const int r1 = r0 + 16;                 // rows of acc1x
    const int i00 = r0 * N + tn0, i01 = r0 * N + tn1;
    const int i10 = r1 * N + tn0, i11 = r1 * N + tn1;
    float v00 = acc00[v], v01 = acc01[v], v10 = acc10[v], v11 = acc11[v];
    if constexpr (HASC) {
      v00 += Cin[i00]; v01 += Cin[i01]; v10 += Cin[i10]; v11 += Cin[i11];
    }
    if constexpr (HASDIAG) {
      if (r0 == tn0) v00 += diag[r0];
      if (r0 == tn1) v01 += diag[r0];
      if (r1 == tn0) v10 += diag[r1];
      if (r1 == tn1) v11 += diag[r1];
    }
    D[i00] = v00; D[i01] = v01; D[i10] = v10; D[i11] = v11;
    if constexpr (HASD2) {
      D2[i00] = v00; D2[i01] = v01; D2[i10] = v10; D2[i11] = v11;
    }
  }
}

// ---------------------------------------------------------------------------
// Init: q = exp(log_q)+eps, r = exp(log_r)+eps, m = 0, V = I.
// ---------------------------------------------------------------------------
__global__ void init_state(const float* __restrict__ log_q,
                           const float* __restrict__ log_r,
                           float* __restrict__ q, float* __restrict__ r,
                           float* __restrict__ m, float* __restrict__ V)
{
  const int idx = blockIdx.x * blockDim.x + threadIdx.x;
  const int i = idx >> 8, j = idx & 255;
  V[idx] = (i == j) ? 1.0f : 0.0f;
  if (idx < N) {
    q[idx] = expf(log_q[idx]) + EPSF;
    r[idx] = expf(log_r[idx]) + EPSF;
    m[idx] = 0.0f;
  }
}

// S = V_pred + diag(r)
__global__ void add_diag_copy(const float* __restrict__ Vp,
                              const float* __restrict__ rr,
                              float* __restrict__ S)
{
  const int idx = blockIdx.x * blockDim.x + threadIdx.x;
  const int i = idx >> 8, j = idx & 255;
  float v = Vp[idx];
  if (i == j) v += rr[i];
  S[idx] = v;
}

// IK = I - K ;  KR = K * diag(r) (column scaling)
__global__ void make_IK_KR(const float* __restrict__ K,
                           const float* __restrict__ rr,
                           float* __restrict__ IK, float* __restrict__ KR)
{
  const int idx = blockIdx.x * blockDim.x + threadIdx.x;
  const int i = idx >> 8, j = idx & 255;
  const float k = K[idx];
  IK[idx] = ((i == j) ? 1.0f : 0.0f) - k;
  KR[idx] = k * rr[j];
}

// m_pred = A @ m  (one block, one row per thread)
__global__ void matvec256(const float* __restrict__ M,
                          const float* __restrict__ x,
                          float* __restrict__ out)
{
  const int rrow = threadIdx.x;
  const float* row = M + (size_t)rrow * N;
  float s = 0.0f;
  for (int c = 0; c < N; ++c) s += row[c] * x[c];
  out[rrow] = s;
}

// In-place right-looking Cholesky on 256x256 S (lower triangle -> L).
__global__ void cholesky256(float* __restrict__ S)
{
  const int tid = threadIdx.x;
  for (int j = 0; j < N; ++j) {
    if (tid == 0) S[j * N + j] = sqrtf(S[j * N + j]);
    __syncthreads();
    const float inv = 1.0f / S[j * N + j];
    const int i = j + 1 + tid;
    if (i < N) S[i * N + j] *= inv;
    __syncthreads();
    if (i < N) {
      const float lij = S[i * N + j];
      for (int k = j + 1; k <= i; ++k) S[i * N + k] -= lij * S[k * N + j];
    }
    __syncthreads();
  }
}

// Linv = L^{-1} (zero above diagonal).  One column per thread.
__global__ void trinv256(const float* __restrict__ L, float* __restrict__ Linv)
{
  const int j = blockIdx.x * blockDim.x + threadIdx.x;
  if (j >= N) return;
  for (int i = 0; i < N; ++i) {
    if (i < j) { Linv[i * N + j] = 0.0f; continue; }
    float s = (i == j) ? 1.0f : 0.0f;
    for (int k = j; k < i; ++k) s -= L[i * N + k] * Linv[k * N + j];
    Linv[i * N + j] = s / L[i * N + i];
  }
}

// Per-step vector math: y, Ky, Sy, w = Linv@y, nis = y.Sy, m_new.
__global__ void kalman_vec(const float* __restrict__ z_t,
                           const float* __restrict__ m_pred,
                           const float* __restrict__ K,
                           const float* __restrict__ Sinv,
                           const float* __restrict__ Linv,
                           float* __restrict__ m_state,
                           float* __restrict__ ms_out,
                           float* __restrict__ ws_out,
                           float* __restrict__ nis_out)
{
  __shared__ float y[N];
  __shared__ float red[N];
  const int r = threadIdx.x;
  const float yr = z_t[r] - m_pred[r];
  y[r] = yr;
  __syncthreads();
  const float* Krow = K    + (size_t)r * N;
  const float* Srow = Sinv + (size_t)r * N;
  const float* Lrow = Linv + (size_t)r * N;
  float ky = 0.0f, sy = 0.0f, wv = 0.0f;
  for (int c = 0; c < N; ++c) {
    const float yc = y[c];
    ky += Krow[c] * yc;
    sy += Srow[c] * yc;
    wv += Lrow[c] * yc;
  }
  const float mn = m_pred[r] + ky;
  m_state[r] = mn;
  ms_out[r]  = mn;
  ws_out[r]  = wv;
  red[r] = yr * sy;
  __syncthreads();
  for (int s = 128; s > 0; s >>= 1) {
    if (r < s) red[r] += red[r + s];
    __syncthreads();
  }
  if (r == 0) nis_out[0] = red[0];
}

// ---------------------------------------------------------------------------
extern "C" void kernel_launch(void* const* d_in, const int* in_sizes, int n_in,
                              void* d_out, int out_size, void* d_ws, size_t ws_size,
                              hipStream_t stream)
{
  (void)in_sizes; (void)n_in; (void)out_size; (void)ws_size;

  const float* obs   = (const float*)d_in[0];   // (1024, 256)
  const float* A     = (const float*)d_in[1];   // (256, 256)
  const float* log_q = (const float*)d_in[2];   // (256,)
  const float* log_r = (const float*)d_in[3];   // (256,)

  float* out     = (float*)d_out;
  float* ms_out  = out;                                   // 1024*256
  float* Vs_out  = out + (size_t)T_STEPS * N;             // 1024*256*256
  float* nis_out = Vs_out + (size_t)T_STEPS * N * N;      // 1024
  float* ws_out  = nis_out + T_STEPS;                     // 1024*256

  float* ws     = (float*)d_ws;
  float* q      = ws;
  float* r      = ws + 256;
  float* m      = ws + 512;
  float* m_pred = ws + 768;
  float* base   = ws + 4096;
  float* V    = base + (size_t)0 * N * N;
  float* Vp   = base + (size_t)1 * N * N;
  float* S    = base + (size_t)2 * N * N;   // Cholesky in place -> L
  float* Linv = base + (size_t)3 * N * N;
  float* Sinv = base + (size_t)4 * N * N;
  float* K    = base + (size_t)5 * N * N;
  float* tmp  = base + (size_t)6 * N * N;
  float* IK   = base + (size_t)7 * N * N;
  float* KR   = base + (size_t)8 * N * N;
  float* tmp2 = base + (size_t)9 * N * N;

  init_state<<<256, 256, 0, stream>>>(log_q, log_r, q, r, m, V);

  for (int t = 0; t < T_STEPS; ++t) {
    // tmp = A @ V
    gemm256<0,0,false,false,false><<<8, 256, 0, stream>>>(A, V, nullptr, nullptr, tmp, nullptr);
    // V_pred = tmp @ A^T + diag(q)
    gemm256<0,1,false,true,false><<<8, 256, 0, stream>>>(tmp, A, nullptr, q, Vp, nullptr);
    // S = V_pred + diag(r)
    add_diag_copy<<<256, 256, 0, stream>>>(Vp, r, S);
    // m_pred = A @ m  (uses previous m; must precede kalman_vec)
    matvec256<<<1, 256, 0, stream>>>(A, m, m_pred);
    // L = chol(S)  (in place)
    cholesky256<<<1, 256, 0, stream>>>(S);
    // Linv = L^{-1}
    trinv256<<<8, 32, 0, stream>>>(S, Linv);
    // Sinv = Linv^T @ Linv
    gemm256<1,0,false,false,false><<<8, 256, 0, stream>>>(Linv, Linv, nullptr, nullptr, Sinv, nullptr);
    // K = V_pred @ Sinv
    gemm256<0,0,false,false,false><<<8, 256, 0, stream>>>(Vp, Sinv, nullptr, nullptr, K, nullptr);
    // y, m_new, nis, w; writes ms/ws/nis outputs and updates m
    kalman_vec<<<1, 256, 0, stream>>>(obs + (size_t)t * N, m_pred, K, Sinv, Linv,
                                      m, ms_out + (size_t)t * N,
                                      ws_out + (size_t)t * N, nis_out + t);
    // IK = I - K ; KR = K * diag(r)
    make_IK_KR<<<256, 256, 0, stream>>>(K, r, IK, KR);
    // tmp2 = IK @ V_pred
    gemm256<0,0,false,false,false><<<8, 256, 0, stream>>>(IK, Vp, nullptr, nullptr, tmp2, nullptr);
    // tmp = tmp2 @ IK^T
    gemm256<0,1,false,false,false><<<8, 256, 0, stream>>>(tmp2, IK, nullptr, nullptr, tmp, nullptr);
    // V_new = KR @ K^T + tmp  -> state V and output Vs[t]
    gemm256<0,1,true,false,true><<<8, 256, 0, stream>>>(KR, K, tmp, nullptr, V,
                                                        Vs_out + (size_t)t * N * N);
  }
}